// VSAETopK_9208409883399
// MI455X (gfx1250) — compile-verified
//
#include <hip/hip_runtime.h>

typedef __attribute__((ext_vector_type(2))) float v2f;
typedef __attribute__((ext_vector_type(4))) float v4f;
typedef __attribute__((ext_vector_type(8))) float v8f;

#define D_ACT       1024
#define D_DICT      32768
#define TOPK        64
#define ROWS_PER_WG 16
#define NTILE       128                  // dict cols per tile: 8 waves x 16
#define NUM_TILES   (D_DICT / NTILE)     // 256

// Fused encoder-GEMM (f32 WMMA) + streaming per-row top-64 + sparse decode.
// One WG = 16 rows of x. Never materializes z (would be 1 GB of traffic).
__global__ __launch_bounds__(256)
void vsae_topk_fused(const float* __restrict__ x,
                     const float* __restrict__ Wenc,
                     const float* __restrict__ benc,
                     const float* __restrict__ Wdec,   // tied: Wdec == Wenc^T
                     const float* __restrict__ bdec,
                     float* __restrict__ out)
{
    __shared__ float zt[ROWS_PER_WG][NTILE + 4];   // z tile (padded vs bank conflicts)
    __shared__ float topa[ROWS_PER_WG][TOPK];      // tracked values (>=0 after relu)
    __shared__ float tops[ROWS_PER_WG][TOPK];      // signed values (== topa here)
    __shared__ int   topi[ROWS_PER_WG][TOPK];      // dict indices
    __shared__ float thr[ROWS_PER_WG];             // running 64th-largest per row

    const int tid  = threadIdx.x;
    const int lane = tid & 31;
    const int wave = tid >> 5;                     // 0..7
    const int row0 = blockIdx.x * ROWS_PER_WG;

    (void)Wdec;  // tied weights: decode reads contiguous Wenc rows instead

    // ---- init top-k state -------------------------------------------------
    for (int i = tid; i < ROWS_PER_WG * TOPK; i += 256) {
        (&topa[0][0])[i] = -1.0f;   // any relu output (>=0) beats this
        (&tops[0][0])[i] = 0.0f;
        (&topi[0][0])[i] = 0;
    }
    if (tid < ROWS_PER_WG) thr[tid] = -1.0f;
    __syncthreads();

    // ---- WMMA f32 16x16x4 fragment coords (ISA 7.12.2) --------------------
    // A 16x4: lanes 0-15 hold (M=lane, K={0,1}); lanes 16-31 (M=lane-16, K={2,3})
    // B 4x16 mirrored: same K-offset split, N = lane&15.
    const int frag_m = lane & 15;
    const int frag_k = (lane >> 4) * 2;

    const float* aBase = x + (size_t)(row0 + frag_m) * D_ACT + frag_k;

    for (int t = 0; t < NUM_TILES; ++t) {
        const int n0 = t * NTILE + wave * 16;      // this wave's 16 dict cols
        const float* bBase = Wenc + (size_t)(n0 + frag_m) * D_ACT + frag_k;

        v8f acc = {};
        #pragma unroll 4
        for (int k = 0; k < D_ACT; k += 4) {
            v2f a = *(const v2f*)(aBase + k);      // 8B load: K pair for this lane
            v2f b = *(const v2f*)(bBase + k);
            acc = __builtin_amdgcn_wmma_f32_16x16x4_f32(
                      false, a, false, b, (short)0, acc, false, false);
        }

        // z = relu(acc + b_enc[n]) -> LDS tile.
        // D layout: VGPR i holds M = i (lanes 0-15) / i+8 (lanes 16-31), N = lane&15.
        const float be     = benc[n0 + frag_m];
        const int   zrow_h = (lane >> 4) * 8;
        #pragma unroll
        for (int i = 0; i < 8; ++i) {
            float z = acc[i] + be;
            zt[i + zrow_h][wave * 16 + frag_m] = z > 0.0f ? z : 0.0f;
        }
        __syncthreads();

        // ---- streaming top-64: wave w owns rows 2w, 2w+1 ------------------
        #pragma unroll
        for (int rr = 0; rr < 2; ++rr) {
            const int row  = wave * 2 + rr;
            float     tcur = thr[row];
            for (int g = 0; g < 4; ++g) {
                const float c = zt[row][g * 32 + lane];
                unsigned mask = (unsigned)__ballot(c > tcur);  // cheap pre-filter
                while (mask) {
                    const int   src = __builtin_ctz(mask);
                    mask &= mask - 1;
                    const float cv  = __shfl(c, src);
                    const int   cig = t * NTILE + g * 32 + src;
                    // argmin over 64 tracked values (2 per lane, wave32)
                    float m0 = topa[row][lane], m1 = topa[row][lane + 32];
                    float m;  int mi;
                    if (m1 < m0) { m = m1; mi = lane + 32; }
                    else         { m = m0; mi = lane; }
                    for (int off = 16; off > 0; off >>= 1) {
                        float om = __shfl_down(m, off);
                        int   oi = __shfl_down(mi, off);
                        if (om < m) { m = om; mi = oi; }
                    }
                    m  = __shfl(m, 0);
                    mi = __shfl(mi, 0);
                    if (cv > m) {                  // fresh check vs current min
                        if (lane == 0) {
                            topa[row][mi] = cv;
                            tops[row][mi] = cv;    // relu: signed == abs
                            topi[row][mi] = cig;
                        }
                        tcur = m;                  // stale-low is still correct
                    }
                }
            }
            // refresh threshold = min of tracked 64
            {
                float m0 = topa[row][lane], m1 = topa[row][lane + 32];
                float m  = m0 < m1 ? m0 : m1;
                for (int off = 16; off > 0; off >>= 1) {
                    float om = __shfl_down(m, off);
                    if (om < m) m = om;
                }
                if (lane == 0) thr[row] = m;
            }
        }
        __syncthreads();   // before zt is overwritten next tile
    }

    // ---- decode: x_hat[r] = sum_j tops[r][j] * Wdec[:, topi[r][j]] + bdec --
    // Tied weights: column d of Wdec == contiguous row d of Wenc -> coalesced,
    // and those rows are L2-hot from the encoder sweep.
    const int a0 = tid * 4;                        // 256 threads x 4 = 1024 acts
    const v4f bd = *(const v4f*)(bdec + a0);
    for (int r = 0; r < ROWS_PER_WG; ++r) {
        v4f accd = bd;
        #pragma unroll 4
        for (int j = 0; j < TOPK; ++j) {
            const float v = tops[r][j];
            const int   d = topi[r][j];
            const v4f   w = *(const v4f*)(Wenc + (size_t)d * D_ACT + a0);
            accd += v * w;
        }
        *(v4f*)(out + (size_t)(row0 + r) * D_ACT + a0) = accd;
    }
}

extern "C" void kernel_launch(void* const* d_in, const int* in_sizes, int n_in,
                              void* d_out, int out_size, void* d_ws, size_t ws_size,
                              hipStream_t stream) {
    const float* x    = (const float*)d_in[0];
    const float* Wenc = (const float*)d_in[1];
    const float* benc = (const float*)d_in[2];
    const float* Wdec = (const float*)d_in[3];
    const float* bdec = (const float*)d_in[4];
    float* out = (float*)d_out;
    (void)n_in; (void)out_size; (void)d_ws; (void)ws_size;

    const int B = in_sizes[0] / D_ACT;             // 8192
    dim3 grid(B / ROWS_PER_WG);                    // 512 WGs, 16 rows each
    dim3 block(256);                               // 8 waves (wave32)
    hipLaunchKernelGGL(vsae_topk_fused, grid, block, 0, stream,
                       x, Wenc, benc, Wdec, bdec, out);
}